// DGCNN_Vanilla_41618233099166
// MI455X (gfx1250) — compile-verified
//
#include <hip/hip_runtime.h>
#include <math.h>

// ---------------------------------------------------------------------------
// DGCNN-vanilla encoder forward, MI455X (gfx1250, wave32).
// fp32 end-to-end; dense GEMMs ride V_WMMA_F32_16X16X4_F32 with the weight
// tile staged into LDS via GLOBAL_LOAD_ASYNC_TO_LDS_B32 (ASYNCcnt path).
// Two-pass training-mode BatchNorm: GEMM(+stats atomics) -> finalize -> norm.
// ---------------------------------------------------------------------------

typedef __attribute__((ext_vector_type(2))) float v2f;
typedef __attribute__((ext_vector_type(8))) float v8f;

#define SLOPE 0.2f
#define EPS_BN 1e-5f

__device__ __forceinline__ float lrelu_f(float a) { return a >= 0.f ? a : SLOPE * a; }

// -------------------------------------------------------------------- zero --
__global__ void zero_kernel(float* __restrict__ p, int n) {
  int i = blockIdx.x * 256 + threadIdx.x;
  if (i < n) p[i] = 0.f;
}

// --------------------------------------------------------------------- kNN --
// grid (N/128, B), block 128. Stage whole batch point cloud (2048x3 f32,
// 24KB) in LDS, each thread = one query point, register top-20 network.
__global__ __launch_bounds__(128) void knn_kernel(const float* __restrict__ x,
                                                  int* __restrict__ idx) {
  __shared__ float xs[2048 * 3];
  const int b = blockIdx.y;
  const int t = threadIdx.x;
  const float* xb = x + (size_t)b * 2048 * 3;
  for (int i = t; i < 2048 * 3; i += 128) xs[i] = xb[i];
  __syncthreads();

  const int q = blockIdx.x * 128 + t;
  const float qx = xs[q * 3], qy = xs[q * 3 + 1], qz = xs[q * 3 + 2];
  const float qq = qx * qx + qy * qy + qz * qz;

  float vals[20];
  int ids[20];
#pragma unroll
  for (int j = 0; j < 20; ++j) { vals[j] = -INFINITY; ids[j] = 0; }

  for (int m = 0; m < 2048; ++m) {
    const float mx = xs[m * 3], my = xs[m * 3 + 1], mz = xs[m * 3 + 2];
    const float mm = mx * mx + my * my + mz * mz;
    float d = 2.f * (qx * mx + qy * my + qz * mz) - qq - mm;  // neg sq dist
    int di = m;
    // branchless stable insertion into descending-sorted list
#pragma unroll
    for (int j = 0; j < 20; ++j) {
      const bool gt = d > vals[j];
      const float tv = gt ? vals[j] : d;
      const int ti = gt ? ids[j] : di;
      vals[j] = gt ? d : vals[j];
      ids[j] = gt ? di : ids[j];
      d = tv; di = ti;
    }
  }
  int* o = idx + ((size_t)b * 2048 + q) * 20;
#pragma unroll
  for (int j = 0; j < 20; ++j) o[j] = ids[j];
}

// ------------------------------------------------- EdgeConv pass 1: stats --
__global__ __launch_bounds__(128) void edge_stats_kernel(
    const float* __restrict__ x, const int* __restrict__ idx,
    const float* __restrict__ W, const float* __restrict__ bias,
    float* __restrict__ sums) {
  __shared__ float ls[128];     // [0,64): sum  [64,128): sumsq
  __shared__ float wsm[6 * 64];
  const int t = threadIdx.x;
  if (t < 128) ls[t] = 0.f;
  for (int i = t; i < 384; i += 128) wsm[i] = W[i];
  __syncthreads();

  const size_t gid = (size_t)blockIdx.x * 128 + t;
  if (gid < (size_t)16 * 2048 * 20) {
    const int b = (int)(gid / (2048 * 20));
    const int r = (int)(gid % (2048 * 20));
    const int n = r / 20;
    const int nb = idx[gid];
    const float* ctr = x + ((size_t)b * 2048 + n) * 3;
    const float* nbr = x + ((size_t)b * 2048 + nb) * 3;
    const float e0 = nbr[0] - ctr[0], e1 = nbr[1] - ctr[1], e2 = nbr[2] - ctr[2];
    const float e3 = ctr[0], e4 = ctr[1], e5 = ctr[2];
    for (int c = 0; c < 64; ++c) {
      float y = bias[c];
      y = fmaf(e0, wsm[c], y);
      y = fmaf(e1, wsm[64 + c], y);
      y = fmaf(e2, wsm[128 + c], y);
      y = fmaf(e3, wsm[192 + c], y);
      y = fmaf(e4, wsm[256 + c], y);
      y = fmaf(e5, wsm[320 + c], y);
      atomicAdd(&ls[c], y);           // ds_add_f32
      atomicAdd(&ls[64 + c], y * y);
    }
  }
  __syncthreads();
  if (t < 128) unsafeAtomicAdd(&sums[t], ls[t]);  // global_atomic_add_f32
}

// ------------------------------------------------------- BN stat finalize --
__global__ void finalize_kernel(const float* __restrict__ sums,
                                const float* __restrict__ g,
                                const float* __restrict__ beta,
                                float* __restrict__ ss, int nC, float invc) {
  const int c = blockIdx.x * 256 + threadIdx.x;
  if (c >= nC) return;
  const float mean = sums[c] * invc;
  const float var = sums[nC + c] * invc - mean * mean;
  const float sc = g[c] * rsqrtf(var + EPS_BN);
  ss[c] = sc;
  ss[nC + c] = beta[c] - mean * sc;
}

// ----------------------------------- EdgeConv pass 2: norm+lrelu+max_k -----
__global__ __launch_bounds__(128) void edge_maxk_kernel(
    const float* __restrict__ x, const int* __restrict__ idx,
    const float* __restrict__ W, const float* __restrict__ bias,
    const float* __restrict__ ss, float* __restrict__ h1) {
  __shared__ float wsm[6 * 64];
  __shared__ float hm[128 * 65];  // padded stride 65 to dodge bank conflicts
  const int t = threadIdx.x;
  for (int i = t; i < 384; i += 128) wsm[i] = W[i];
  float* my = hm + t * 65;
  for (int c = 0; c < 64; ++c) my[c] = -INFINITY;
  __syncthreads();

  const int gid = blockIdx.x * 128 + t;  // b*2048+n, < 32768
  const int b = gid >> 11;
  const float* ctr = x + (size_t)gid * 3;
  const int* id = idx + (size_t)gid * 20;
  for (int k = 0; k < 20; ++k) {
    const float* nbr = x + ((size_t)b * 2048 + id[k]) * 3;
    const float e0 = nbr[0] - ctr[0], e1 = nbr[1] - ctr[1], e2 = nbr[2] - ctr[2];
    const float e3 = ctr[0], e4 = ctr[1], e5 = ctr[2];
    for (int c = 0; c < 64; ++c) {
      float y = bias[c];
      y = fmaf(e0, wsm[c], y);
      y = fmaf(e1, wsm[64 + c], y);
      y = fmaf(e2, wsm[128 + c], y);
      y = fmaf(e3, wsm[192 + c], y);
      y = fmaf(e4, wsm[256 + c], y);
      y = fmaf(e5, wsm[320 + c], y);
      const float a = lrelu_f(fmaf(y, ss[c], ss[64 + c]));
      my[c] = fmaxf(my[c], a);
    }
  }
  float* o = h1 + (size_t)gid * 64;
  for (int c = 0; c < 64; ++c) o[c] = my[c];
}

// -------------------------------------------------- WMMA fp32 GEMM + bias --
// Block = 256 threads (8 waves) computes a 128(M) x 64(N) output tile.
// Weight K-chunk (KC x 64, <=32KB) is staged into LDS with
// GLOBAL_LOAD_ASYNC_TO_LDS_B32 (ASYNCcnt) and shared by all 8 waves.
// Each wave owns 16 rows x 64 cols = 4 accumulators; per K-step one A
// fragment (2 global lane-loads) feeds 4 V_WMMA_F32_16X16X4_F32 whose B
// fragments come from LDS. Compute predicate (row0 < M) is wave-uniform so
// EXEC is all-ones inside WMMA. Optional per-column sum/sumsq for BN.
__global__ __launch_bounds__(256) void wmma_gemm_kernel(
    const float* __restrict__ A, const float* __restrict__ W,
    const float* __restrict__ bias, float* __restrict__ Y,
    float* __restrict__ sums, int M, int Kd, int Nout) {
  __shared__ float Wt[128 * 64];  // 32 KB K-chunk of weights
  const int t = threadIdx.x;
  const int wave = t >> 5;
  const int lane = t & 31;
  const int hi = lane >> 4;   // 0: lanes 0-15, 1: lanes 16-31
  const int l16 = lane & 15;
  const int row0 = blockIdx.x * 128 + wave * 16;
  const int colBase = blockIdx.y * 64;
  const bool active = (row0 < M);
  const int KC = (Kd >= 128) ? 128 : Kd;

  v8f acc[4] = {{}, {}, {}, {}};

  for (int kc0 = 0; kc0 < Kd; kc0 += KC) {
    // ---- async stage of W[kc0:kc0+KC, colBase:colBase+64) into LDS ----
    for (int i = t; i < KC * 64; i += 256) {
      const int kk = i >> 6, cc = i & 63;
      const unsigned lds = (unsigned)(uintptr_t)(&Wt[i]);
      const unsigned long long ga =
          (unsigned long long)(uintptr_t)(W + (size_t)(kc0 + kk) * Nout + colBase + cc);
      asm volatile("global_load_async_to_lds_b32 %0, %1, off"
                   :: "v"(lds), "v"(ga) : "memory");
    }
    asm volatile("s_wait_asynccnt 0" ::: "memory");
    __syncthreads();

    if (active) {
      const float* arow = A + (size_t)(row0 + l16) * Kd + kc0;
      if (kc0 + KC < Kd) __builtin_prefetch(arow + KC, 0, 1);  // global_prefetch_b8
      for (int k = 0; k < KC; k += 4) {
        const int ak = k + (hi << 1);
        v2f a;
        a.x = arow[ak];
        a.y = arow[ak + 1];
#pragma unroll
        for (int cg = 0; cg < 4; ++cg) {
          v2f b;
          b.x = Wt[ak * 64 + cg * 16 + l16];
          b.y = Wt[(ak + 1) * 64 + cg * 16 + l16];
#if __has_builtin(__builtin_amdgcn_wmma_f32_16x16x4_f32)
          acc[cg] = __builtin_amdgcn_wmma_f32_16x16x4_f32(
              false, a, false, b, (short)0, acc[cg], false, false);
#else
          acc[cg][0] = fmaf(a.x, b.x, acc[cg][0]);  // fallback (builtin absent)
          acc[cg][1] = fmaf(a.y, b.y, acc[cg][1]);
#endif
        }
      }
    }
    __syncthreads();  // protect Wt before next chunk overwrite
  }

  if (!active) return;
#pragma unroll
  for (int cg = 0; cg < 4; ++cg) {
    const int col = colBase + cg * 16 + l16;
    const float bv = bias[col];
    float s = 0.f, s2 = 0.f;
#pragma unroll
    for (int r = 0; r < 8; ++r) {
      const int row = row0 + (hi << 3) + r;
      const float v = acc[cg][r] + bv;
      Y[(size_t)row * Nout + col] = v;
      s += v;
      s2 = fmaf(v, v, s2);
    }
    if (sums) {
      s += __shfl_down(s, 16);
      s2 += __shfl_down(s2, 16);
      if (hi == 0) {
        unsafeAtomicAdd(&sums[col], s);
        unsafeAtomicAdd(&sums[Nout + col], s2);
      }
    }
  }
}

// ----------------------------------------------------- elementwise BN+act --
__global__ void bn_act_kernel(const float* __restrict__ y,
                              const float* __restrict__ ss,
                              float* __restrict__ act, int nC, int total) {
  const int i = blockIdx.x * 256 + threadIdx.x;
  if (i >= total) return;
  const int c = i % nC;
  act[i] = lrelu_f(fmaf(y[i], ss[c], ss[nC + c]));
}

// -------------------------------- fused BN+lrelu + max over 2048 points ----
__global__ __launch_bounds__(256) void colmax_bn_kernel(
    const float* __restrict__ y, const float* __restrict__ ss,
    float* __restrict__ pooled) {
  const int c = blockIdx.x * 256 + threadIdx.x;  // < 1024
  const int b = blockIdx.y;
  const float* yb = y + (size_t)b * 2048 * 1024;
  const float sc = ss[c], sh = ss[1024 + c];
  float m = -INFINITY;
  for (int n = 0; n < 2048; ++n) {
    const float a = lrelu_f(fmaf(yb[(size_t)n * 1024 + c], sc, sh));
    m = fmaxf(m, a);
  }
  pooled[b * 1024 + c] = m;
}

// ---------------------------------------------------------------------------
extern "C" void kernel_launch(void* const* d_in, const int* in_sizes, int n_in,
                              void* d_out, int out_size, void* d_ws,
                              size_t ws_size, hipStream_t stream) {
  (void)in_sizes; (void)n_in; (void)out_size; (void)ws_size;
  const float* x = (const float*)d_in[0];
  const float* eW = (const float*)d_in[1];
  const float* eB = (const float*)d_in[2];
  const float* eG = (const float*)d_in[3];
  const float* eBeta = (const float*)d_in[4];
  const float* pwW[3] = {(const float*)d_in[5], (const float*)d_in[6], (const float*)d_in[7]};
  const float* pwB[3] = {(const float*)d_in[8], (const float*)d_in[9], (const float*)d_in[10]};
  const float* pwG[3] = {(const float*)d_in[11], (const float*)d_in[12], (const float*)d_in[13]};
  const float* pwBeta[3] = {(const float*)d_in[14], (const float*)d_in[15], (const float*)d_in[16]};
  const float* liW[2] = {(const float*)d_in[17], (const float*)d_in[18]};
  const float* liB[2] = {(const float*)d_in[19], (const float*)d_in[20]};
  const float* liG[2] = {(const float*)d_in[21], (const float*)d_in[22]};
  const float* liBeta[2] = {(const float*)d_in[23], (const float*)d_in[24]};
  const float* oW = (const float*)d_in[25];
  const float* oB = (const float*)d_in[26];
  float* out = (float*)d_out;

  char* ws = (char*)d_ws;
  size_t off = 0;
  auto take = [&](size_t bytes) -> void* {
    void* p = ws + off;
    off += (bytes + 255) & ~(size_t)255;
    return p;
  };
  int* idx = (int*)take((size_t)16 * 2048 * 20 * 4);        // kNN indices
  float* sums = (float*)take(4736 * 4);                     // BN sum/sumsq
  float* ssb = (float*)take(4736 * 4);                      // BN scale/shift
  float* h1 = (float*)take((size_t)32768 * 64 * 4);         // EdgeConv out
  float* ybuf = (float*)take((size_t)32768 * 1024 * 4);     // pre-activation
  float* act = (float*)take((size_t)32768 * 128 * 4);       // activations
  float* pooled = (float*)take((size_t)16 * 1024 * 4);
  float* la1 = (float*)take((size_t)16 * 512 * 4);
  float* la2 = (float*)take((size_t)16 * 512 * 4);

  // per-layer offsets (floats) inside sums/ssb
  const int S_EDGE = 0, S_PW1 = 128, S_PW2 = 384, S_PW3 = 640,
            S_LIN1 = 2688, S_LIN2 = 3712;

  zero_kernel<<<(4736 + 255) / 256, 256, 0, stream>>>(sums, 4736);

  // kNN graph
  knn_kernel<<<dim3(16, 16), 128, 0, stream>>>(x, idx);

  // EdgeConv (K=6 GEMM, VALU) -> BN stats -> finalize -> norm+lrelu+max_k
  edge_stats_kernel<<<5120, 128, 0, stream>>>(x, idx, eW, eB, sums + S_EDGE);
  finalize_kernel<<<1, 256, 0, stream>>>(sums + S_EDGE, eG, eBeta, ssb + S_EDGE,
                                         64, 1.f / (16.f * 2048.f * 20.f));
  edge_maxk_kernel<<<256, 128, 0, stream>>>(x, idx, eW, eB, ssb + S_EDGE, h1);

  // pw1: [32768,64] @ [64,128]
  wmma_gemm_kernel<<<dim3(256, 2), 256, 0, stream>>>(h1, pwW[0], pwB[0], ybuf,
                                                     sums + S_PW1, 32768, 64, 128);
  finalize_kernel<<<1, 256, 0, stream>>>(sums + S_PW1, pwG[0], pwBeta[0],
                                         ssb + S_PW1, 128, 1.f / 32768.f);
  bn_act_kernel<<<(32768 * 128) / 256, 256, 0, stream>>>(ybuf, ssb + S_PW1, act,
                                                         128, 32768 * 128);
  // pw2: [32768,128] @ [128,128]
  wmma_gemm_kernel<<<dim3(256, 2), 256, 0, stream>>>(act, pwW[1], pwB[1], ybuf,
                                                     sums + S_PW2, 32768, 128, 128);
  finalize_kernel<<<1, 256, 0, stream>>>(sums + S_PW2, pwG[1], pwBeta[1],
                                         ssb + S_PW2, 128, 1.f / 32768.f);
  bn_act_kernel<<<(32768 * 128) / 256, 256, 0, stream>>>(ybuf, ssb + S_PW2, act,
                                                         128, 32768 * 128);
  // pw3: [32768,128] @ [128,1024], then fused BN+lrelu+max over points
  wmma_gemm_kernel<<<dim3(256, 16), 256, 0, stream>>>(act, pwW[2], pwB[2], ybuf,
                                                      sums + S_PW3, 32768, 128, 1024);
  finalize_kernel<<<4, 256, 0, stream>>>(sums + S_PW3, pwG[2], pwBeta[2],
                                         ssb + S_PW3, 1024, 1.f / 32768.f);
  colmax_bn_kernel<<<dim3(4, 16), 256, 0, stream>>>(ybuf, ssb + S_PW3, pooled);

  // lin1: [16,1024] @ [1024,512]
  wmma_gemm_kernel<<<dim3(1, 8), 256, 0, stream>>>(pooled, liW[0], liB[0], ybuf,
                                                   sums + S_LIN1, 16, 1024, 512);
  finalize_kernel<<<2, 256, 0, stream>>>(sums + S_LIN1, liG[0], liBeta[0],
                                         ssb + S_LIN1, 512, 1.f / 16.f);
  bn_act_kernel<<<(16 * 512) / 256, 256, 0, stream>>>(ybuf, ssb + S_LIN1, la1,
                                                      512, 16 * 512);
  // lin2: [16,512] @ [512,512]
  wmma_gemm_kernel<<<dim3(1, 8), 256, 0, stream>>>(la1, liW[1], liB[1], ybuf,
                                                   sums + S_LIN2, 16, 512, 512);
  finalize_kernel<<<2, 256, 0, stream>>>(sums + S_LIN2, liG[1], liBeta[1],
                                         ssb + S_LIN2, 512, 1.f / 16.f);
  bn_act_kernel<<<(16 * 512) / 256, 256, 0, stream>>>(ybuf, ssb + S_LIN2, la2,
                                                      512, 16 * 512);
  // out: [16,512] @ [512,1024] + bias -> d_out (no BN)
  wmma_gemm_kernel<<<dim3(1, 16), 256, 0, stream>>>(la2, oW, oB, out, nullptr,
                                                    16, 512, 1024);
}